// SupConLoss_90117003805311
// MI455X (gfx1250) — compile-verified
//
#include <hip/hip_runtime.h>
#include <hip/hip_bf16.h>
#include <math.h>
#include <stdint.h>

// SupCon loss fused on gfx1250 (MI455X, wave32).
// sim = F@F^T/T via V_WMMA_F32_16X16X4_F32; column panels double-buffered in
// LDS via GLOBAL_LOAD_ASYNC_TO_LDS_B64 (ASYNCcnt), stored pre-swizzled into
// WMMA B-fragment order so feeds are bank-conflict-minimal ds_load_b128.

typedef float v2f __attribute__((ext_vector_type(2)));
typedef float v4f __attribute__((ext_vector_type(4)));
typedef float v8f __attribute__((ext_vector_type(8)));
typedef int   v2i __attribute__((ext_vector_type(2)));

typedef __attribute__((address_space(1))) v2i g_v2i;   // global 64-bit chunk
typedef __attribute__((address_space(3))) v2i l_v2i;   // LDS 64-bit chunk

#define BATCH        8192
#define DIM          128
#define INV_T        10.0f            // 1 / 0.1
#define KSTEPS       (DIM / 4)        // 32 WMMA k-steps per 16x16 tile
#define COL_TILES    (BATCH / 16)     // 512
#define WAVES_PER_WG 8
#define ROWS_PER_WG  (WAVES_PER_WG * 16)    // 128
#define PANEL_TILES  2                       // column tiles staged per panel
#define NPANELS      (COL_TILES / PANEL_TILES)  // 256
#define LANE_STRIDE  272               // 256B fragment + 16B pad (bank spread, 16B align)
#define TILE_BYTES   (32 * LANE_STRIDE)          // 8704
#define PANEL_BYTES  (PANEL_TILES * TILE_BYTES)  // 17408 (x2 buffers = 34816 B LDS)

#if defined(__has_builtin)
#if __has_builtin(__builtin_amdgcn_global_load_async_to_lds_b64)
#define HAVE_ASYNC_LDS 1
#endif
#endif
#ifndef HAVE_ASYNC_LDS
#define HAVE_ASYNC_LDS 0
#endif

__device__ __forceinline__ void async_cp_b64(const char* gsrc, char* ldst) {
#if HAVE_ASYNC_LDS
    __builtin_amdgcn_global_load_async_to_lds_b64(
        (g_v2i*)(uintptr_t)gsrc,
        (l_v2i*)(uint32_t)(uintptr_t)ldst,
        0, 0);
#else
    *(uint64_t*)ldst = *(const uint64_t*)gsrc;   // global_load_b64 + ds_store_b64
#endif
}

__device__ __forceinline__ void wait_async_all() {
#if HAVE_ASYNC_LDS
#if __has_builtin(__builtin_amdgcn_s_wait_asynccnt)
    __builtin_amdgcn_s_wait_asynccnt(0);
#else
    asm volatile("s_wait_asynccnt 0x0" ::: "memory");
#endif
#endif
}

// Stage one panel (PANEL_TILES 16-column tiles) into LDS, pre-swizzled into
// B-fragment order: the 8 bytes at offset t*TILE_BYTES + L*LANE_STRIDE + k*8 of the panel buffer hold
// F[col0 + 16t + L%16][4k + 2*(L/16) + {0,1}].
// Each async b64 copy: lane l handles (h = l&1, k = l>>1); global side of one
// instruction reads a contiguous 256B row segment (fully coalesced).
__device__ __forceinline__ void stage_panel(const float* __restrict__ F, int col0,
                                            char* buf, int wave, int lane)
{
#pragma unroll
    for (int i = 0; i < 4; ++i) {                  // 32 (tile,row) pairs / 8 waves
        const int pr = wave * 4 + i;
        const int t  = pr >> 4;                    // tile in panel
        const int rr = pr & 15;                    // row within tile
        const char* src = (const char*)(F + (size_t)(col0 + 16 * t + rr) * DIM)
                        + (lane >> 1) * 16 + (lane & 1) * 8;
        char* dst = buf + t * TILE_BYTES
                  + (rr + 16 * (lane & 1)) * LANE_STRIDE + (lane >> 1) * 8;
        async_cp_b64(src,        dst);             // k = 0..15
        async_cp_b64(src + 256,  dst + 128);       // k = 16..31
    }
}

__global__ __launch_bounds__(256)
void supcon_rows_kernel(const float* __restrict__ F,
                        const int*   __restrict__ labels,
                        float* __restrict__ row_val,
                        float* __restrict__ row_cnt)
{
    __shared__ __align__(16) char lds[2 * PANEL_BYTES];

    const int tid  = threadIdx.x;
    const int lane = tid & 31;
    const int wave = tid >> 5;
    const int half = lane >> 4;
    const int l16  = lane & 15;
    const int row0 = (blockIdx.x * WAVES_PER_WG + wave) * 16;

    // ---- A fragment: rows row0..row0+15, full K, resident in VGPRs ----
    v2f afrag[KSTEPS];
    {
        const v2f* arow = (const v2f*)(F + (size_t)(row0 + l16) * DIM);
#pragma unroll
        for (int k = 0; k < KSTEPS; ++k)
            afrag[k] = arow[2 * k + half];
    }

    int rlab[8];
#pragma unroll
    for (int r = 0; r < 8; ++r)
        rlab[r] = labels[row0 + r + 8 * half];

    float m[8], s[8], ps[8], pc[8];
#pragma unroll
    for (int r = 0; r < 8; ++r) { m[r] = -INFINITY; s[r] = 0.0f; ps[r] = 0.0f; pc[r] = 0.0f; }

    // prologue: stage panel 0
    stage_panel(F, 0, lds, wave, lane);
    wait_async_all();
    __syncthreads();

    for (int p = 0; p < NPANELS; ++p) {
        char* cur = lds + ((p)     & 1) * PANEL_BYTES;
        char* nxt = lds + ((p + 1) & 1) * PANEL_BYTES;

        if (p + 1 < NPANELS)                       // async: overlaps with compute below
            stage_panel(F, (p + 1) * (PANEL_TILES * 16), nxt, wave, lane);

        if (p + 2 < NPANELS) {                     // L2 prefetch of panel p+2
            const char* pf = (const char*)(F + (size_t)((p + 2) * (PANEL_TILES * 16)
                                                        + (tid & 31)) * DIM)
                           + (tid >> 5) * 64;
            __builtin_prefetch(pf, 0, 1);
        }

#pragma unroll
        for (int t = 0; t < PANEL_TILES; ++t) {
            const int   col0 = p * (PANEL_TILES * 16) + t * 16;
            const char* frag = cur + t * TILE_BYTES + lane * LANE_STRIDE;

            v8f c = {0.0f, 0.0f, 0.0f, 0.0f, 0.0f, 0.0f, 0.0f, 0.0f};
#pragma unroll
            for (int hk = 0; hk < 2; ++hk) {       // split K to bound VGPR pressure
                v4f q[8];
#pragma unroll
                for (int j = 0; j < 8; ++j)        // 8x ds_load_b128
                    q[j] = *(const v4f*)(frag + hk * 128 + j * 16);
#pragma unroll
                for (int j = 0; j < 8; ++j) {
                    const v2f b0 = {q[j].x, q[j].y};
                    const v2f b1 = {q[j].z, q[j].w};
                    c = __builtin_amdgcn_wmma_f32_16x16x4_f32(
                            false, afrag[hk * 16 + 2 * j],     false, b0, (short)0, c, false, false);
                    c = __builtin_amdgcn_wmma_f32_16x16x4_f32(
                            false, afrag[hk * 16 + 2 * j + 1], false, b1, (short)0, c, false, false);
                }
            }

            const int jcol = col0 + l16;
            const int clab = labels[jcol];
#pragma unroll
            for (int r = 0; r < 8; ++r) {
                const int   irow = row0 + r + 8 * half;
                const float x    = c[r] * INV_T;
                const bool  diag = (jcol == irow);

                const float mn = fmaxf(m[r], x);
                float e = __expf(x - mn);
                if (diag) e = 0.0f;
                s[r] = fmaf(s[r], __expf(m[r] - mn), e);
                m[r] = mn;

                if ((clab == rlab[r]) && !diag) { ps[r] += x; pc[r] += 1.0f; }
            }
        }

        wait_async_all();      // panel p+1 fully in LDS
        __syncthreads();       // everyone done reading cur before it is re-staged
    }

    // ---- reduce across the 16 column slots (lanes within each half) ----
#pragma unroll
    for (int r = 0; r < 8; ++r) {
        float mr = m[r], sr = s[r], pr = ps[r], cr = pc[r];
#pragma unroll
        for (int off = 1; off < 16; off <<= 1) {
            const float m2 = __shfl_xor(mr, off, 32);
            const float s2 = __shfl_xor(sr, off, 32);
            const float p2 = __shfl_xor(pr, off, 32);
            const float c2 = __shfl_xor(cr, off, 32);
            const float mn = fmaxf(mr, m2);
            sr = sr * __expf(mr - mn) + s2 * __expf(m2 - mn);
            mr = mn;
            pr += p2;
            cr += c2;
        }
        if (l16 == 0) {
            const int   irow  = row0 + r + 8 * half;
            const float lse   = __logf(sr + 1e-12f);
            const float mlp   = (pr - cr * (mr + lse)) / fmaxf(cr, 1.0f);
            const bool  valid = (cr > 0.0f);
            row_val[irow] = valid ? -mlp : 0.0f;
            row_cnt[irow] = valid ? 1.0f : 0.0f;
        }
    }
}

__global__ __launch_bounds__(256)
void supcon_reduce_kernel(const float* __restrict__ row_val,
                          const float* __restrict__ row_cnt,
                          float* __restrict__ out)
{
    __shared__ float sv[256];
    __shared__ float sc[256];
    const int tid = threadIdx.x;
    float v = 0.0f, n = 0.0f;
    for (int i = tid; i < BATCH; i += 256) {
        v += row_val[i];
        n += row_cnt[i];
    }
    sv[tid] = v;
    sc[tid] = n;
    __syncthreads();
    for (int off = 128; off > 0; off >>= 1) {
        if (tid < off) { sv[tid] += sv[tid + off]; sc[tid] += sc[tid + off]; }
        __syncthreads();
    }
    if (tid == 0) {
        const float nv = sc[0];
        out[0] = (nv > 0.0f) ? (sv[0] / nv) : 0.0f;
    }
}

extern "C" void kernel_launch(void* const* d_in, const int* in_sizes, int n_in,
                              void* d_out, int out_size, void* d_ws, size_t ws_size,
                              hipStream_t stream) {
    (void)in_sizes; (void)n_in; (void)out_size; (void)ws_size;
    const float* F      = (const float*)d_in[0];
    const int*   labels = (const int*)d_in[1];

    float* row_val = (float*)d_ws;          // BATCH floats
    float* row_cnt = row_val + BATCH;       // BATCH floats (64 KB total)

    supcon_rows_kernel<<<dim3(BATCH / ROWS_PER_WG), dim3(256), 0, stream>>>(
        F, labels, row_val, row_cnt);
    supcon_reduce_kernel<<<dim3(1), dim3(256), 0, stream>>>(
        row_val, row_cnt, (float*)d_out);
}